// encoderT_36747740184884
// MI455X (gfx1250) — compile-verified
//
#include <hip/hip_runtime.h>

typedef float v2f __attribute__((ext_vector_type(2)));
typedef float v8f __attribute__((ext_vector_type(8)));

// ---------------- degree / norm ----------------

__global__ void init_deg_kernel(float* __restrict__ deg, int n) {
  int i = blockIdx.x * blockDim.x + threadIdx.x;
  if (i < n) deg[i] = 1.0f;  // self-loop contribution
}

__global__ void count_deg_kernel(const int* __restrict__ dst, float* __restrict__ deg, int E) {
  int e = blockIdx.x * blockDim.x + threadIdx.x;
  if (e < E) atomicAdd(&deg[dst[e]], 1.0f);
}

__global__ void rsqrt_kernel(float* __restrict__ d, int n) {
  int i = blockIdx.x * blockDim.x + threadIdx.x;
  if (i < n) d[i] = rsqrtf(d[i]);  // deg >= 1 always (self-loops)
}

// ---------------- GEMM: g = (X @ W) * dinv[row]; also AGG = g (self-loop seed) ----
// One wave computes a 16x16 f32 tile with V_WMMA_F32_16X16X4_F32, K-loop CIN/4.
// Block = 64 threads (2 waves); wave w covers column tile (blockIdx.y*2 + w).
// CIN and COUT are compile-time so all global addresses are base+imm offsets.

template <int CIN, int COUT>
__global__ void gemm_scale_kernel(const float* __restrict__ X,
                                  const float* __restrict__ W,
                                  const float* __restrict__ dinv,
                                  float* __restrict__ G,
                                  float* __restrict__ AGG,
                                  int N) {
  const int lane = threadIdx.x & 31;
  const int wave = threadIdx.x >> 5;
  const int m    = lane & 15;   // row/col within tile for this lane
  const int hi   = lane >> 4;   // half-wave selector
  const int row0 = blockIdx.x * 16;
  const int col0 = (blockIdx.y * 2 + wave) * 16;

  // Clamp load row to keep EXEC all-ones for WMMA (stores are predicated).
  const int lrow = min(row0 + m, N - 1);
  const float* __restrict__ xrow = X + (size_t)lrow * CIN + 2 * hi;
  const float* __restrict__ wcol = W + col0 + m + (size_t)(2 * hi) * COUT;

  v8f acc = {};
#pragma unroll
  for (int k0 = 0; k0 < CIN; k0 += 4) {
    // A (16x4 f32): VGPR0 = K=ka, VGPR1 = K=ka+1 for this lane's row (ka = k0+2*hi)
    v2f a;
    a.x = xrow[k0];
    a.y = xrow[k0 + 1];
    // B (4x16 f32): VGPR0 = row ka, VGPR1 = row ka+1, column col0+m
    v2f b;
    b.x = wcol[(size_t)k0 * COUT];
    b.y = wcol[(size_t)(k0 + 1) * COUT];
    acc = __builtin_amdgcn_wmma_f32_16x16x4_f32(
        /*neg_a=*/false, a, /*neg_b=*/false, b,
        /*c_mod=*/(short)0, acc, /*reuse_a=*/false, /*reuse_b=*/false);
  }

  // C/D layout: VGPR r holds rows (r, r+8) split across lane halves.
#pragma unroll
  for (int r = 0; r < 8; ++r) {
    const int row = row0 + r + hi * 8;
    if (row < N) {
      const float v = acc[r] * dinv[row];
      const size_t idx = (size_t)row * COUT + col0 + m;
      G[idx] = v;    // scaled features for edge gather
      AGG[idx] = v;  // self-loop seed for scatter accumulation
    }
  }
}

// ---------------- edge scatter: AGG[dst] += G[src] (pure add, norm pre-folded) ----

__global__ void scatter_add_kernel(const int* __restrict__ src,
                                   const int* __restrict__ dst,
                                   const float* __restrict__ G,
                                   float* __restrict__ AGG,
                                   int E, int C) {
  const int cq = C >> 2;  // float4 chunks per node
  const long long t = (long long)blockIdx.x * blockDim.x + threadIdx.x;
  const long long e = t / cq;
  const int q = (int)(t - e * cq);
  if (e >= E) return;
  // hint: pull upcoming edge indices toward the cache hierarchy
  __builtin_prefetch(src + e + 2048, 0, 0);
  __builtin_prefetch(dst + e + 2048, 0, 0);
  const int s = src[e];
  const int d = dst[e];
  const float4 v = ((const float4*)(G + (size_t)s * C))[q];
  float* ap = AGG + (size_t)d * C + (size_t)q * 4;
  atomicAdd(ap + 0, v.x);
  atomicAdd(ap + 1, v.y);
  atomicAdd(ap + 2, v.z);
  atomicAdd(ap + 3, v.w);
}

// ---------------- epilogue: out = AGG * dinv[row] + bias, optional ReLU ----------

__global__ void epilogue_kernel(const float* __restrict__ AGG,
                                const float* __restrict__ dinv,
                                const float* __restrict__ bias,
                                float* __restrict__ OUT,
                                int n, int C, int do_relu) {
  const long long i = (long long)blockIdx.x * blockDim.x + threadIdx.x;
  const long long total = (long long)n * C;
  if (i >= total) return;
  const int row = (int)(i / C);
  const int f = (int)(i - (long long)row * C);
  float v = AGG[i] * dinv[row] + bias[f];
  if (do_relu) v = fmaxf(v, 0.0f);
  OUT[i] = v;
}

// ---------------- driver ----------------

extern "C" void kernel_launch(void* const* d_in, const int* in_sizes, int n_in,
                              void* d_out, int out_size, void* d_ws, size_t ws_size,
                              hipStream_t stream) {
  const float* x  = (const float*)d_in[0];
  const int* ei   = (const int*)d_in[1];
  const float* W1 = (const float*)d_in[2];
  const float* b1 = (const float*)d_in[3];
  const float* W2 = (const float*)d_in[4];
  const float* b2 = (const float*)d_in[5];
  const float* W3 = (const float*)d_in[6];
  const float* b3 = (const float*)d_in[7];

  const int C_IN = 64, C_HID = 64, C_OUT = 32;
  const int N = in_sizes[0] / C_IN;  // 100000
  const int E = in_sizes[1] / 2;     // 1000000
  const int* src = ei;
  const int* dst = ei + E;

  float* ws = (float*)d_ws;
  float* dinv = ws;                            // N
  float* bufA = ws + N;                        // N*64 : g (scaled GEMM out)
  float* bufB = bufA + (size_t)N * 64;         // N*64 : agg
  float* bufC = bufB + (size_t)N * 64;         // N*64 : next-layer input

  const int TB = 256;
  const dim3 blk1((N + TB - 1) / TB);
  const dim3 blkE((E + TB - 1) / TB);

  // symmetric norm: dinv = rsqrt(1 + in-degree)
  init_deg_kernel<<<blk1, TB, 0, stream>>>(dinv, N);
  count_deg_kernel<<<blkE, TB, 0, stream>>>(dst, dinv, E);
  rsqrt_kernel<<<blk1, TB, 0, stream>>>(dinv, N);

  const int rowBlocks = (N + 15) / 16;

  // ---- layer 1: 64 -> 64, ReLU ----
  gemm_scale_kernel<64, 64><<<dim3(rowBlocks, C_HID / 32), 64, 0, stream>>>(
      x, W1, dinv, bufA, bufB, N);
  {
    long long work = (long long)E * (C_HID / 4);
    scatter_add_kernel<<<(int)((work + TB - 1) / TB), TB, 0, stream>>>(
        src, dst, bufA, bufB, E, C_HID);
  }
  {
    long long total = (long long)N * C_HID;
    epilogue_kernel<<<(int)((total + TB - 1) / TB), TB, 0, stream>>>(
        bufB, dinv, b1, bufC, N, C_HID, 1);
  }

  // ---- layer 2: 64 -> 64, ReLU ----
  gemm_scale_kernel<64, 64><<<dim3(rowBlocks, C_HID / 32), 64, 0, stream>>>(
      bufC, W2, dinv, bufA, bufB, N);
  {
    long long work = (long long)E * (C_HID / 4);
    scatter_add_kernel<<<(int)((work + TB - 1) / TB), TB, 0, stream>>>(
        src, dst, bufA, bufB, E, C_HID);
  }
  {
    long long total = (long long)N * C_HID;
    epilogue_kernel<<<(int)((total + TB - 1) / TB), TB, 0, stream>>>(
        bufB, dinv, b2, bufC, N, C_HID, 1);
  }

  // ---- layer 3: 64 -> 32, no ReLU, write d_out ----
  gemm_scale_kernel<64, 32><<<dim3(rowBlocks, C_OUT / 32), 64, 0, stream>>>(
      bufC, W3, dinv, bufA, bufB, N);
  {
    long long work = (long long)E * (C_OUT / 4);
    scatter_add_kernel<<<(int)((work + TB - 1) / TB), TB, 0, stream>>>(
        src, dst, bufA, bufB, E, C_OUT);
  }
  {
    long long total = (long long)N * C_OUT;
    epilogue_kernel<<<(int)((total + TB - 1) / TB), TB, 0, stream>>>(
        bufB, dinv, b3, (float*)d_out, N, C_OUT, 0);
  }
}